// GCNConv_52140902974209
// MI455X (gfx1250) — compile-verified
//
#include <hip/hip_runtime.h>
#include <hip/hip_bf16.h>
#include <math.h>

typedef float v2f __attribute__((ext_vector_type(2)));
typedef float v8f __attribute__((ext_vector_type(8)));

#define DFEAT 128

// ---------------- degree / norm ----------------

__global__ __launch_bounds__(256) void deg_init_kernel(float* __restrict__ deg, int n) {
    int i = blockIdx.x * 256 + threadIdx.x;
    if (i < n) deg[i] = 1.0f;   // self-loop contributes 1 to out-degree
}

__global__ __launch_bounds__(256) void deg_count_kernel(const int* __restrict__ src,
                                                        float* __restrict__ deg, int e) {
    int i = blockIdx.x * 256 + threadIdx.x;
    if (i < e) atomicAdd(&deg[src[i]], 1.0f);
}

__global__ __launch_bounds__(256) void deg_rsqrt_kernel(float* __restrict__ deg, int n) {
    int i = blockIdx.x * 256 + threadIdx.x;
    if (i < n) {
        float d = deg[i];
        deg[i] = (d > 0.0f) ? rsqrtf(d) : 0.0f;   // becomes norm[]
    }
}

// ---------------- h = x @ W via V_WMMA_F32_16X16X4_F32 ----------------
// Block: 256 threads = 8 waves. Each wave computes a 16(M) x 128(N) strip of h.
// W is staged transposed into LDS once per block: ldsWt[n][k], so a B-fragment
// (two consecutive K values for a fixed N) is one aligned ds_load_b64.

__global__ __launch_bounds__(256) void gemm_wmma_kernel(const float* __restrict__ x,
                                                        const float* __restrict__ W,
                                                        float* __restrict__ h, int n) {
    __shared__ float ldsWt[DFEAT * DFEAT];   // 64 KB of the WGP's 320 KB LDS

    // cooperative load + transpose of W (row-major k x n -> ldsWt[n*128 + k])
    for (int i = threadIdx.x; i < DFEAT * DFEAT; i += 256) {
        int k  = i >> 7;
        int nn = i & 127;
        ldsWt[nn * DFEAT + k] = W[i];
    }
    __syncthreads();

    const int wave = threadIdx.x >> 5;
    const int lane = threadIdx.x & 31;
    const int m0 = (blockIdx.x * 8 + wave) * 16;
    if (m0 >= n) return;                      // whole-wave exit: EXEC stays all-1s for WMMA

    const int row   = lane & 15;              // M (for A) / N (for B) within tile
    const int khalf = (lane >> 4) * 2;        // lanes 16-31 hold K+2,K+3

    v8f acc[8] = {};                          // 16x128 strip = 8 accumulators

    const float* xrow = x + (size_t)(m0 + row) * DFEAT + khalf;

    for (int kk = 0; kk < DFEAT; kk += 4) {
        v2f a = *(const v2f*)(xrow + kk);     // global_load_b64, 8B aligned
#pragma unroll
        for (int j = 0; j < 8; ++j) {
            const int nidx = j * 16 + row;
            v2f b = *(const v2f*)&ldsWt[nidx * DFEAT + kk + khalf];  // ds_load_b64
            acc[j] = __builtin_amdgcn_wmma_f32_16x16x4_f32(
                false, a, false, b, (short)0, acc[j], false, false);
        }
    }

    // C/D layout: VGPR r, lanes 0-15 -> M=r, lanes 16-31 -> M=r+8; N = lane&15
    const int mbase = m0 + (lane >> 4) * 8;
    const int nbase = lane & 15;
#pragma unroll
    for (int j = 0; j < 8; ++j) {
#pragma unroll
        for (int r = 0; r < 8; ++r) {
            h[(size_t)(mbase + r) * DFEAT + j * 16 + nbase] = acc[j][r];
        }
    }
}

// ---------------- agg init with self-loop message ----------------
// agg[i] = h[i] * norm[i]  (the i->i self-loop message); also fully
// initializes d_out (harness poisons it).

__global__ __launch_bounds__(256) void init_agg_kernel(const float* __restrict__ h,
                                                       const float* __restrict__ norm,
                                                       float* __restrict__ agg, int n) {
    int t = blockIdx.x * 256 + threadIdx.x;
    int node = t >> 5;
    if (node >= n) return;
    int f = (t & 31) * 4;
    float ns = norm[node];
    float4 v = *(const float4*)&h[(size_t)node * DFEAT + f];
    v.x *= ns; v.y *= ns; v.z *= ns; v.w *= ns;
    *(float4*)&agg[(size_t)node * DFEAT + f] = v;
}

// ---------------- per-edge scatter: agg[dst] += h[src]*norm[src] ----------------
// 32 lanes per edge, float4 gather (h resident in 192MB L2), 4x f32 atomics.

__global__ __launch_bounds__(256) void scatter_kernel(const int* __restrict__ src,
                                                      const int* __restrict__ dst,
                                                      const float* __restrict__ h,
                                                      const float* __restrict__ norm,
                                                      float* __restrict__ agg, int e) {
    int t = blockIdx.x * 256 + threadIdx.x;
    int edge = t >> 5;
    if (edge >= e) return;
    int f = (t & 31) * 4;
    int s = src[edge];
    int d = dst[edge];
    float ns = norm[s];
    float4 v = *(const float4*)&h[(size_t)s * DFEAT + f];
    float* o = &agg[(size_t)d * DFEAT + f];
    atomicAdd(o + 0, v.x * ns);
    atomicAdd(o + 1, v.y * ns);
    atomicAdd(o + 2, v.z * ns);
    atomicAdd(o + 3, v.w * ns);
}

// ---------------- finalize: *norm[dst], L2 row-normalize, tanh ----------------
// One wave32 per node; float4 per lane; wave shuffle reduction for ||row||_2.

__global__ __launch_bounds__(256) void finalize_kernel(const float* __restrict__ norm,
                                                       float* __restrict__ out, int n) {
    int t = blockIdx.x * 256 + threadIdx.x;
    int node = t >> 5;
    if (node >= n) return;
    int lane = t & 31;
    float nd = norm[node];
    float4 v = *(const float4*)&out[(size_t)node * DFEAT + lane * 4];
    v.x *= nd; v.y *= nd; v.z *= nd; v.w *= nd;
    float ss = v.x * v.x + v.y * v.y + v.z * v.z + v.w * v.w;
#pragma unroll
    for (int off = 16; off > 0; off >>= 1)
        ss += __shfl_xor(ss, off, 32);
    float inv = 1.0f / fmaxf(sqrtf(ss), 1e-12f);
    float4 r;
    r.x = tanhf(v.x * inv);
    r.y = tanhf(v.y * inv);
    r.z = tanhf(v.z * inv);
    r.w = tanhf(v.w * inv);
    *(float4*)&out[(size_t)node * DFEAT + lane * 4] = r;
}

// ---------------- launcher ----------------

extern "C" void kernel_launch(void* const* d_in, const int* in_sizes, int n_in,
                              void* d_out, int out_size, void* d_ws, size_t ws_size,
                              hipStream_t stream) {
    const float* x  = (const float*)d_in[0];
    const int*   ei = (const int*)d_in[1];
    const float* W  = (const float*)d_in[2];
    float* out = (float*)d_out;

    const int n = in_sizes[0] / DFEAT;   // 100000 nodes
    const int e = in_sizes[1] / 2;       // 625000 edges
    const int* src = ei;                 // edge_index[0]
    const int* dst = ei + e;             // edge_index[1]

    // workspace layout: [norm: n floats][h: n*128 floats]  (~51.6 MB)
    float* norm = (float*)d_ws;
    float* h    = norm + n;

    deg_init_kernel <<<(n + 255) / 256, 256, 0, stream>>>(norm, n);
    deg_count_kernel<<<(e + 255) / 256, 256, 0, stream>>>(src, norm, e);
    deg_rsqrt_kernel<<<(n + 255) / 256, 256, 0, stream>>>(norm, n);

    // 8 waves/block, each wave = 16 rows -> 128 rows per block
    gemm_wmma_kernel<<<(n + 127) / 128, 256, 0, stream>>>(x, W, h, n);

    // 32 lanes per node/edge for the feature dimension (float4 each)
    long nt_node = (long)n * 32;
    long nt_edge = (long)e * 32;
    init_agg_kernel<<<(int)((nt_node + 255) / 256), 256, 0, stream>>>(h, norm, out, n);
    scatter_kernel <<<(int)((nt_edge + 255) / 256), 256, 0, stream>>>(src, dst, h, norm, out, e);
    finalize_kernel<<<(int)((nt_node + 255) / 256), 256, 0, stream>>>(norm, out, n);
}